// QDense_64115271795093
// MI455X (gfx1250) — compile-verified
//
#include <hip/hip_runtime.h>
#include <math.h>
#include <stdint.h>

#define BATCH 512
#define DDIM  256
#define ODIM  128

typedef __attribute__((ext_vector_type(2))) float v2f;
typedef __attribute__((ext_vector_type(4))) float v4f;
typedef __attribute__((ext_vector_type(8))) float v8f;
typedef __attribute__((ext_vector_type(4))) int   v4i;

#if defined(__HIP_DEVICE_COMPILE__)
#if !__has_builtin(__builtin_amdgcn_wmma_f32_16x16x4_f32)
#error "missing __builtin_amdgcn_wmma_f32_16x16x4_f32 on device toolchain"
#endif
#if __has_builtin(__builtin_amdgcn_global_load_async_to_lds_b128) && \
    __has_builtin(__builtin_amdgcn_s_wait_asynccnt)
#define HAVE_ASYNC 1
#else
#define HAVE_ASYNC 0
#endif
#else
#define HAVE_ASYNC 0
#endif

typedef __attribute__((address_space(1))) v4i* gptr_v4i;
typedef __attribute__((address_space(3))) v4i* lptr_v4i;

// ---------------------------------------------------------------------------
// LDS layout (floats):
//   RI : [8 waves][2 (R,I)][16 rows][260]  = 66560 floats (row stride 260 ->
//        conflict-free b64 A-fragment reads in stage 2)
//   XS : 2 ping-pong buffers x [2 (r,i)][128 rows][16 cols] = 8192 floats
//   PE : [128] exp(rand_w);  PS : [1] sum;  TR : [8] trace partials
// total = 74889 floats = 299556 B  (< 320 KB WGP LDS)
// ---------------------------------------------------------------------------
#define RI_OFF 0
#define XS_OFF 66560
#define PE_OFF (66560 + 8192)
#define PS_OFF (PE_OFF + 128)
#define TR_OFF (PS_OFF + 1)
#define SMEM_FLOATS (TR_OFF + 8)

__device__ __forceinline__ v8f wmma_f32(v2f a, v2f b, v8f c) {
#if defined(__HIP_DEVICE_COMPILE__)
    return __builtin_amdgcn_wmma_f32_16x16x4_f32(false, a, false, b,
                                                 (short)0, c, false, false);
#else
    return c;
#endif
}

// 16-byte copy global -> LDS. Async (ASYNCcnt / VGPR-free) when available.
__device__ __forceinline__ void cp16(const float* g, float* l) {
#if HAVE_ASYNC
    __builtin_amdgcn_global_load_async_to_lds_b128(
        (gptr_v4i)(uintptr_t)g,
        (lptr_v4i)(uint32_t)(uintptr_t)l,
        0, 0);
#else
    *(v4f*)l = *(const v4f*)g;
#endif
}

template <int N>
__device__ __forceinline__ void async_wait_le() {
#if HAVE_ASYNC
    __builtin_amdgcn_s_wait_asynccnt(N);      // immediate must be a constant
#endif
}

__global__ void __launch_bounds__(256)
qdense_fused_kernel(const float* __restrict__ x_real,
                    const float* __restrict__ x_imag,
                    const float* __restrict__ weight,   // [256][256][2] interleaved r,i
                    const float* __restrict__ rand_w,   // [128]
                    const float* __restrict__ lam,      // [1]
                    float* __restrict__ out)            // [2][512][128][128]
{
    extern __shared__ float smem[];
    float* RI = smem + RI_OFF;
    float* XS = smem + XS_OFF;
    float* PE = smem + PE_OFF;
    float* PS = smem + PS_OFF;
    float* TR = smem + TR_OFF;

    const int b    = blockIdx.x;
    const int t    = threadIdx.x;
    const int lane = t & 31;
    const int m    = t >> 5;        // wave id 0..7 -> owns output rows 16m..16m+15
    const int hi   = lane >> 4;     // half-wave select (K split in WMMA layouts)
    const int c    = lane & 15;     // lane within half

    // --- softmax(rand_w) prep (tiny, once per block) ------------------------
    if (t < ODIM) PE[t] = __expf(rand_w[t]);
    __syncthreads();
    if (t == 0) { float s = 0.f; for (int i = 0; i < ODIM; ++i) s += PE[i]; PS[0] = s; }
    const float sgm = 1.f / (1.f + __expf(-lam[0]));

    const float* xr0 = x_real + (size_t)b * DDIM * DDIM;
    const float* xi0 = x_imag + (size_t)b * DDIM * DDIM;

    // =======================================================================
    // Stage 1:  R = U_r x_r - U_i x_i ,  I = U_i x_r + U_r x_i  (rows 0..127)
    // 32 half-strips (128 rows x 16 cols), double-buffered async into LDS.
    // =======================================================================
    const int rowA = 16 * m + c;                 // U row (< 128) for A-fragments
    float* baseR = RI + (size_t)(m * 2 + 0) * 16 * 260;
    float* baseI = RI + (size_t)(m * 2 + 1) * 16 * 260;

    const int pmat = t >> 7;                     // this thread stages real(0)/imag(1)
    const int prow = t & 127;                    // local row it stages

    auto strip_prefetch = [&](int j) {           // stage half-strip j -> buf[j&1]
        const int nj = j >> 1, hj = j & 1;
        const float* src = (pmat ? xi0 : xr0) +
                           (size_t)(128 * hj + prow) * DDIM + nj * 16;
        float* dst = XS + (j & 1) * 4096 + pmat * 2048 + prow * 16;
        cp16(src + 0,  dst + 0);
        cp16(src + 4,  dst + 4);
        cp16(src + 8,  dst + 8);
        cp16(src + 12, dst + 12);                // 4 async ops / thread / half
    };

    strip_prefetch(0);
    v8f accR = {}; v8f accI = {};
    for (int it = 0; it < 32; ++it) {
        const int n  = it >> 1;                  // column strip 0..15
        const int hh = it & 1;                   // which 128-row half
        __syncthreads();                         // everyone done reading buf[it+1 & 1]
        if (it + 1 < 32) {
            strip_prefetch(it + 1);              // overlap DMA with compute below
            async_wait_le<4>();                  // oldest 4 (buf[it&1]) retired
        } else {
            async_wait_le<0>();                  // final half: drain everything
        }
        __syncthreads();                         // buf[it&1] globally visible

        const float* B = XS + (it & 1) * 4096;
        #pragma unroll 4
        for (int k = 0; k < 128; k += 4) {
            const int kl = k + 2 * hi;           // local row in buffer
            const int kg = 128 * hh + kl;        // global K for weight column
            // one b128 pulls A-fragments of BOTH U_r and U_i (interleaved layout)
            v4f w = *(const v4f*)(weight + ((size_t)rowA * DDIM + kg) * 2);
            v2f aUr = {w.x, w.z};
            v2f aUi = {w.y, w.w};
            const float* pr = B + kl * 16 + c;
            const float* pi = B + 2048 + kl * 16 + c;
            v2f bXr  = { pr[0],  pr[16] };
            v2f bXi  = { pi[0],  pi[16] };
            v2f bXiN = { -bXi.x, -bXi.y };       // f32 WMMA has no A/B neg bits
            accR = wmma_f32(aUr, bXr,  accR);
            accR = wmma_f32(aUi, bXiN, accR);
            accI = wmma_f32(aUi, bXr,  accI);
            accI = wmma_f32(aUr, bXi,  accI);
        }
        if (hh) {                                // strip n complete -> park in LDS
            #pragma unroll
            for (int v = 0; v < 8; ++v) {
                const int r = v + 8 * hi;        // C-layout row
                baseR[r * 260 + 16 * n + c] = accR[v];
                baseI[r * 260 + 16 * n + c] = accI[v];
            }
            accR = v8f{}; accI = v8f{};
        }
    }

    // =======================================================================
    // Stage 2:  out_r = R U_r^T + I U_i^T ,  out_i = I U_r^T - R U_i^T
    //           A-fragments from LDS (b64), B-fragments from weight (b128).
    // =======================================================================
    const float* rBase = baseR + c * 260;        // local row = lane%16
    const float* iBase = baseI + c * 260;

    auto compute_tile = [&](int n2, v8f& Or, v8f& Oi) {
        Or = v8f{}; Oi = v8f{};
        const int rowB = 16 * n2 + c;            // U row (<128) giving B = U^T frag
        #pragma unroll 4
        for (int k = 0; k < DDIM; k += 4) {
            const int kk = k + 2 * hi;
            v2f aR = *(const v2f*)(rBase + kk);
            v2f aI = *(const v2f*)(iBase + kk);
            v4f w  = *(const v4f*)(weight + ((size_t)rowB * DDIM + kk) * 2);
            v2f bUr  = { w.x,  w.z};
            v2f bUi  = { w.y,  w.w};
            v2f bUiN = {-w.y, -w.w};
            Or = wmma_f32(aR, bUr,  Or);
            Or = wmma_f32(aI, bUi,  Or);
            Oi = wmma_f32(aI, bUr,  Oi);
            Oi = wmma_f32(aR, bUiN, Oi);
        }
    };

    // Diagonal tile (n2 == m) first: needed for the trace.
    v8f dOr, dOi;
    compute_tile(m, dOr, dOi);
    {
        float part = 0.f;                        // diag lives on lanes v and v+24
        #pragma unroll
        for (int v = 0; v < 8; ++v) {
            if (lane == v)      part += dOr[v];
            if (lane == v + 24) part += dOr[v];
        }
        #pragma unroll
        for (int off = 16; off > 0; off >>= 1) part += __shfl_xor(part, off);
        if (lane == 0) TR[m] = part;
    }
    __syncthreads();                             // all traces (and PS) ready

    float tr = 0.f;
    #pragma unroll
    for (int w8 = 0; w8 < 8; ++w8) tr += TR[w8];
    const float inv    = sgm / tr;
    const float oneMs  = 1.f - sgm;
    const float invPS  = 1.f / PS[0];

    float* o_r = out + (size_t)b * ODIM * ODIM;
    float* o_i = out + (size_t)BATCH * ODIM * ODIM + (size_t)b * ODIM * ODIM;

    auto write_tile = [&](int n2, const v8f& Or, const v8f& Oi) {
        #pragma unroll
        for (int v = 0; v < 8; ++v) {
            const int i = 16 * m + v + 8 * hi;
            const int j = 16 * n2 + c;
            float vr = Or[v] * inv;
            if (i == j) vr += oneMs * PE[i] * invPS;   // + (1-s)*diag(softmax)
            o_r[(size_t)i * ODIM + j] = vr;
            o_i[(size_t)i * ODIM + j] = Oi[v] * inv;
        }
    };

    write_tile(m, dOr, dOi);
    for (int n2 = 0; n2 < 8; ++n2) {
        if (n2 == m) continue;
        v8f Or, Oi;
        compute_tile(n2, Or, Oi);
        write_tile(n2, Or, Oi);
    }
}

extern "C" void kernel_launch(void* const* d_in, const int* in_sizes, int n_in,
                              void* d_out, int out_size, void* d_ws, size_t ws_size,
                              hipStream_t stream) {
    (void)in_sizes; (void)n_in; (void)out_size; (void)d_ws; (void)ws_size;
    const float* x_real = (const float*)d_in[0];
    const float* x_imag = (const float*)d_in[1];
    const float* weight = (const float*)d_in[2];
    const float* rand_w = (const float*)d_in[3];
    const float* lam    = (const float*)d_in[4];
    float* out = (float*)d_out;

    const size_t shmem = (size_t)SMEM_FLOATS * sizeof(float);   // ~299.6 KB
    qdense_fused_kernel<<<dim3(BATCH), dim3(256), shmem, stream>>>(
        x_real, x_imag, weight, rand_w, lam, out);
}